// Model_9216999817958
// MI455X (gfx1250) — compile-verified
//
#include <hip/hip_runtime.h>
#include <hip/hip_bf16.h>

typedef float v2f __attribute__((ext_vector_type(2)));
typedef float v8f __attribute__((ext_vector_type(8)));

#define HID 128

// ---------------- elementwise / utility kernels ----------------

__global__ void kfill(float* __restrict__ p, float v, int n) {
  int i = blockIdx.x * blockDim.x + threadIdx.x;
  if (i < n) p[i] = v;
}

__global__ void kcount(const int* __restrict__ idx, float* __restrict__ deg, int n) {
  int i = blockIdx.x * blockDim.x + threadIdx.x;
  if (i < n) atomicAdd(&deg[idx[i]], 1.0f);
}

__global__ void krsqrt(float* __restrict__ d, int n) {
  int i = blockIdx.x * blockDim.x + threadIdx.x;
  if (i < n) d[i] = rsqrtf(fmaxf(d[i], 1.0f));
}

// out[row][f] = h[row][f] * r[row]
__global__ void kscale(const float* __restrict__ h, const float* __restrict__ r,
                       float* __restrict__ o, int total) {
  int i = blockIdx.x * blockDim.x + threadIdx.x;
  if (i < total) o[i] = h[i] * r[i >> 7];
}

__global__ void kaddbias(float* __restrict__ o, const float* __restrict__ b, int total) {
  int i = blockIdx.x * blockDim.x + threadIdx.x;
  if (i < total) o[i] += b[i & 127];
}

// h = prelu(hc * (g*rsqrt(1+eps)) + bb, alpha); g/bb nullable
__global__ void kbnprelu(float* __restrict__ o, const float* __restrict__ hc,
                         const float* __restrict__ g, const float* __restrict__ bb,
                         const float* __restrict__ alpha, int total) {
  int i = blockIdx.x * blockDim.x + threadIdx.x;
  if (i >= total) return;
  float y = hc[i];
  if (g) y = y * (g[i & 127] * rsqrtf(1.0f + 1e-5f)) + bb[i & 127];
  float a = alpha[0];
  o[i] = (y >= 0.0f) ? y : a * y;
}

// ---------------- WMMA f32 GEMM ----------------
// C[M x N] = A[M x K] * op(B) (+ bias per column)
// BT==0: B is K x N row-major.  BT==1: B is N x K row-major (use B^T).
// One wave computes a 16x32 slab (2 WMMA tiles sharing the A fragment);
// a 128-thread block covers 16x128. Out-of-range rows/cols need NO operand
// masking: A row m only feeds C row m, B col n only feeds C col n, and those
// C elements are skipped by the store guards. Pointers are clamped so every
// load is unconditional (global_load_b64, no EXEC juggling). BT is a template
// parameter so the inner loop is branch-free straight-line code.
template <int BT>
__global__ __launch_bounds__(128) void kgemm(const float* __restrict__ A,
                                             const float* __restrict__ B,
                                             const float* __restrict__ bias,
                                             float* __restrict__ C,
                                             int M, int N, int K) {
  int wave = threadIdx.x >> 5;
  int lane = threadIdx.x & 31;
  int r = lane & 15;
  int hi = lane >> 4;
  int tm = blockIdx.y << 4;
  int tn0 = (blockIdx.x << 7) + (wave << 5);
  if (tn0 >= N) return;                 // uniform per wave
  int tn1 = tn0 + 16;

  int arow = tm + r;
  bool aok = arow < M;
  int bc0 = tn0 + r; bool b0ok = bc0 < N;
  int bc1 = tn1 + r; bool b1ok = bc1 < N;

  const float* Ap = A + (size_t)(aok ? arow : 0) * K + (hi << 1);
  const float* B0;
  const float* B1;
  if (BT) {
    B0 = B + (size_t)(b0ok ? bc0 : 0) * K + (hi << 1);
    B1 = B + (size_t)(b1ok ? bc1 : 0) * K + (hi << 1);
  } else {
    B0 = B + (size_t)(hi << 1) * N + (b0ok ? bc0 : 0);
    B1 = B + (size_t)(hi << 1) * N + (b1ok ? bc1 : 0);
  }

  v8f acc0 = {0.f, 0.f, 0.f, 0.f, 0.f, 0.f, 0.f, 0.f};
  v8f acc1 = {0.f, 0.f, 0.f, 0.f, 0.f, 0.f, 0.f, 0.f};
#pragma unroll 8
  for (int k0 = 0; k0 < K; k0 += 4) {
    v2f a = *(const v2f*)Ap;
    v2f b0, b1;
    if (BT) {
      b0 = *(const v2f*)B0;
      b1 = *(const v2f*)B1;
      B0 += 4; B1 += 4;
    } else {
      b0.x = B0[0]; b0.y = B0[N];
      b1.x = B1[0]; b1.y = B1[N];
      B0 += (size_t)4 * N; B1 += (size_t)4 * N;
    }
    Ap += 4;
    acc0 = __builtin_amdgcn_wmma_f32_16x16x4_f32(false, a, false, b0, (short)0, acc0,
                                                 false, false);
    acc1 = __builtin_amdgcn_wmma_f32_16x16x4_f32(false, a, false, b1, (short)0, acc1,
                                                 false, false);
  }

  if (b0ok) {
    float bv = bias ? bias[bc0] : 0.0f;
#pragma unroll
    for (int v = 0; v < 8; ++v) {
      int crow = tm + v + (hi << 3);
      if (crow < M) C[(size_t)crow * N + bc0] = acc0[v] + bv;
    }
  }
  if (b1ok) {
    float bv = bias ? bias[bc1] : 0.0f;
#pragma unroll
    for (int v = 0; v < 8; ++v) {
      int crow = tm + v + (hi << 3);
      if (crow < M) C[(size_t)crow * N + bc1] = acc1[v] + bv;
    }
  }
}

// scatter: out[dst[e]] += x[src[e]] * rdi[dst[e]]   (128 feats, 4 per thread)
__global__ void kscatter(const float* __restrict__ x, const int* __restrict__ src,
                         const int* __restrict__ dst, const float* __restrict__ rdi,
                         float* __restrict__ out, int nE) {
  int gid = blockIdx.x * blockDim.x + threadIdx.x;
  int e = gid >> 5;
  if (e >= nE) return;
  int f = (gid & 31) << 2;
  int s = src[e], d = dst[e];
  float sc = rdi[d];
  const float* xp = x + (size_t)s * 128 + f;
  float* op = out + (size_t)d * 128 + f;
  atomicAdd(op + 0, xp[0] * sc);
  atomicAdd(op + 1, xp[1] * sc);
  atomicAdd(op + 2, xp[2] * sc);
  atomicAdd(op + 3, xp[3] * sc);
}

// ---------------- semantic attention ----------------
// S[k] += sum_n sum_h tanh(t[n][h]) * w2[h]
__global__ void ktanhdot(const float* __restrict__ t, const float* __restrict__ w2,
                         float* __restrict__ S, int k, int N) {
  __shared__ float red[256];
  float local = 0.f;
  for (int n = blockIdx.x * blockDim.x + threadIdx.x; n < N;
       n += gridDim.x * blockDim.x) {
    const float* tp = t + (size_t)n * 128;
    float s = 0.f;
#pragma unroll 4
    for (int h = 0; h < 128; ++h) s += tanhf(tp[h]) * w2[h];
    local += s;
  }
  red[threadIdx.x] = local;
  __syncthreads();
  for (int off = 128; off > 0; off >>= 1) {
    if ((int)threadIdx.x < off) red[threadIdx.x] += red[threadIdx.x + off];
    __syncthreads();
  }
  if (threadIdx.x == 0) atomicAdd(&S[k], red[0]);
}

__global__ void ksoftmax(const float* __restrict__ S, float* __restrict__ beta,
                         int K, float invN) {
  if (blockIdx.x || threadIdx.x) return;
  float m = -1e30f;
  for (int k = 0; k < K; ++k) m = fmaxf(m, S[k] * invN);
  float e[8], sum = 0.f;
  for (int k = 0; k < K; ++k) { e[k] = expf(S[k] * invN - m); sum += e[k]; }
  for (int k = 0; k < K; ++k) beta[k] = e[k] / sum;
}

__global__ void kcombine(const float* __restrict__ z0, const float* __restrict__ z1,
                         const float* __restrict__ z2, const float* __restrict__ z3,
                         const float* __restrict__ beta, float* __restrict__ o,
                         int K, int total) {
  int i = blockIdx.x * blockDim.x + threadIdx.x;
  if (i >= total) return;
  float v = beta[0] * z0[i];
  if (K > 1) v += beta[1] * z1[i];
  if (K > 2) v += beta[2] * z2[i];
  if (K > 3) v += beta[3] * z3[i];
  o[i] = v;
}

// ---------------- GAT ----------------
__global__ void keler(const float* __restrict__ z, const float* __restrict__ al,
                      const float* __restrict__ ar, float* __restrict__ el,
                      float* __restrict__ er, int N) {
  int gid = blockIdx.x * blockDim.x + threadIdx.x;
  if (gid >= N * 2) return;
  int n = gid >> 1, h = gid & 1;
  const float* zp = z + (size_t)n * 256 + h * 128;
  const float* alp = al + h * 128;
  const float* arp = ar + h * 128;
  float a = 0.f, b = 0.f;
#pragma unroll 4
  for (int f = 0; f < 128; ++f) { a += zp[f] * alp[f]; b += zp[f] * arp[f]; }
  el[gid] = a;
  er[gid] = b;
}

__global__ void kbuildedges(const int* __restrict__ s, const int* __restrict__ d,
                            int* __restrict__ gs, int* __restrict__ gd,
                            int so, int dof, int eoff, int nE) {
  int i = blockIdx.x * blockDim.x + threadIdx.x;
  if (i >= nE) return;
  gs[eoff + i] = s[i] + so;
  gd[eoff + i] = d[i] + dof;
}

__device__ __forceinline__ void atomicMaxFloat(float* addr, float val) {
  unsigned int* ua = (unsigned int*)addr;
  unsigned int cur = *ua;
  while (__uint_as_float(cur) < val) {
    unsigned int assumed = cur;
    cur = atomicCAS(ua, assumed, __float_as_uint(val));
    if (cur == assumed) break;
  }
}

__global__ void kedgemax(const int* __restrict__ gs, const int* __restrict__ gd,
                         const float* __restrict__ el, const float* __restrict__ er,
                         float* __restrict__ eattr, float* __restrict__ m, int nE) {
  int gid = blockIdx.x * blockDim.x + threadIdx.x;
  if (gid >= nE * 2) return;
  int e = gid >> 1, h = gid & 1;
  float v = el[gs[e] * 2 + h] + er[gd[e] * 2 + h];
  v = (v >= 0.0f) ? v : 0.2f * v;           // leaky_relu 0.2
  eattr[gid] = v;
  atomicMaxFloat(&m[gd[e] * 2 + h], v);
}

__global__ void kedgeexp(const int* __restrict__ gd, const float* __restrict__ m,
                         float* __restrict__ eattr, float* __restrict__ den, int nE) {
  int gid = blockIdx.x * blockDim.x + threadIdx.x;
  if (gid >= nE * 2) return;
  int e = gid >> 1, h = gid & 1;
  float a = expf(eattr[gid] - m[gd[e] * 2 + h]);
  eattr[gid] = a;
  atomicAdd(&den[gd[e] * 2 + h], a);
}

__global__ void kedgeapply(const int* __restrict__ gs, const int* __restrict__ gd,
                           const float* __restrict__ eattr, const float* __restrict__ den,
                           const float* __restrict__ z, float* __restrict__ outg, int nE) {
  long long gid = (long long)blockIdx.x * blockDim.x + threadIdx.x;
  int e = (int)(gid >> 6);
  if (e >= nE) return;
  int c = (int)(gid & 63);
  int f = c << 2;            // 0..252 over both heads
  int h = f >> 7;
  int s = gs[e], d = gd[e];
  float alpha = eattr[e * 2 + h] / den[d * 2 + h];
  const float* zp = z + (size_t)s * 256 + f;
  float* op = outg + (size_t)d * 256 + f;
  atomicAdd(op + 0, alpha * zp[0]);
  atomicAdd(op + 1, alpha * zp[1]);
  atomicAdd(op + 2, alpha * zp[2]);
  atomicAdd(op + 3, alpha * zp[3]);
}

__global__ void kgatpost(const float* __restrict__ outg, const float* __restrict__ bias,
                         const float* __restrict__ ag, const float* __restrict__ ao,
                         float* __restrict__ o, int N) {
  int gid = blockIdx.x * blockDim.x + threadIdx.x;
  if (gid >= N * 128) return;
  int n = gid >> 7, f = gid & 127;
  float a = ag[0], b = ao[0];
  float v0 = outg[(size_t)n * 256 + f] + bias[f];
  float v1 = outg[(size_t)n * 256 + 128 + f] + bias[128 + f];
  v0 = (v0 >= 0.f) ? v0 : a * v0;
  v1 = (v1 >= 0.f) ? v1 : a * v1;
  float mn = 0.5f * (v0 + v1);
  o[gid] = (mn >= 0.f) ? mn : b * mn;
}

// ---------------- host orchestration ----------------

extern "C" void kernel_launch(void* const* d_in, const int* in_sizes, int n_in,
                              void* d_out, int out_size, void* d_ws, size_t ws_size,
                              hipStream_t stream) {
  (void)in_sizes; (void)n_in; (void)out_size; (void)ws_size;

  const float* P = (const float*)d_in[0];
  // node types: 0 disease, 1 drug, 2 gene, 3 pathway, 4 protein
  static const int NT_N[5]   = {3000, 2000, 8000, 1500, 8000};
  static const int NT_OFF[5] = {0, 3000, 5000, 13000, 14500};
  const int NTOT = 22500;
  const float* X[5] = {(const float*)d_in[1], (const float*)d_in[2],
                       (const float*)d_in[3], (const float*)d_in[4],
                       (const float*)d_in[5]};
  // relations in ALL_RELS order
  static const int RSRC[10] = {1, 1, 0, 1, 4, 4, 2, 2, 3, 3};
  static const int RDST[10] = {1, 0, 0, 4, 4, 2, 2, 3, 3, 0};
  static const int RNE[10]  = {30000, 60000, 30000, 80000, 250000,
                               150000, 250000, 50000, 15000, 40000};
  static const int SORTIDX[10] = {2, 1, 0, 3, 9, 8, 4, 5, 7, 6}; // rel -> sorted conv slot

  // param offsets (floats), jax pytree sorted-key flatten order
  const size_t P_DECW = 0;
  const size_t P_FG1 = 16384, P_FG2 = 181761;          // bn_b+0, bn_g+128, convs+256, prelu+165376
  const size_t P_GAT = 347138;                          // W+0, al+32768, ar+33024, bias+33280, pg+33536, po+33537
  const size_t P_LATT_DIS = 380676, P_LATT_DRUG = 397316; // W1+0, b1+16384, w2+16512
  const size_t P_LIN = 413956;                          // per ntype stride 65664: W+0, b+65536
  const size_t P_SATT = 742276;                         // W1+0, b1+16384, w2+16512
  static const size_t BLK_BASE[5] = {758916, 808453, 907527, 857990, 957064}; // dd, dp, pg, gpw, pwdis

  // workspace carve
  float* W = (float*)d_ws;
  size_t off = 0;
  auto alloc = [&](size_t n) { float* p = W + off; off += n; return p; };
  float* h[5]; float* hc[5]; float* zA[5]; float* zB[5];
  for (int t = 0; t < 5; ++t) h[t]  = alloc((size_t)NT_N[t] * HID);
  for (int t = 0; t < 5; ++t) hc[t] = alloc((size_t)NT_N[t] * HID);
  for (int t = 0; t < 5; ++t) zA[t] = alloc((size_t)NT_N[t] * HID);
  for (int t = 0; t < 5; ++t) zB[t] = alloc((size_t)NT_N[t] * HID);
  float* tmp  = alloc(8000 * HID);
  float* xbuf = alloc(8000 * HID);
  float* dego = alloc(8000);
  float* degi = alloc(8000);
  float* drugL[4]; float* disL[4];
  for (int k = 0; k < 4; ++k) drugL[k] = alloc(2000 * HID);
  for (int k = 0; k < 4; ++k) disL[k]  = alloc(3000 * HID);
  float* Sbuf = alloc(8);
  float* Beta = alloc(8);
  float* zg   = alloc((size_t)NTOT * 256);
  float* el   = alloc(NTOT * 2);
  float* er   = alloc(NTOT * 2);
  float* mbuf = alloc(NTOT * 2);
  float* den  = alloc(NTOT * 2);
  float* outg = alloc((size_t)NTOT * 256);
  float* gout = alloc((size_t)NTOT * HID);
  const int ETOT = 955000;
  float* eattr = alloc((size_t)ETOT * 2);
  int* geS = (int*)alloc(ETOT);
  int* geD = (int*)alloc(ETOT);
  float* Rbuf = alloc(2000 * HID);
  float* Dmat = alloc(3000 * HID);
  float* Dd   = alloc(3000 * HID);

  auto nb = [](long long n) { return (unsigned)((n + 255) / 256); };
  auto gemm = [&](const float* A, const float* B, const float* bias, float* C,
                  int M, int N, int K, int bt) {
    dim3 g((N + 127) / 128, (M + 15) / 16);
    if (bt)
      kgemm<1><<<g, 128, 0, stream>>>(A, B, bias, C, M, N, K);
    else
      kgemm<0><<<g, 128, 0, stream>>>(A, B, bias, C, M, N, K);
  };

  auto run_rel = [&](int r, const float* Wc, const float* bc) {
    int s = RSRC[r], d = RDST[r];
    int Ns = NT_N[s], Nd = NT_N[d], nE = RNE[r];
    const int* es = (const int*)d_in[6 + 2 * r];
    const int* ed = (const int*)d_in[7 + 2 * r];
    kfill<<<nb(Ns), 256, 0, stream>>>(dego, 0.f, Ns);
    kfill<<<nb(Nd), 256, 0, stream>>>(degi, 0.f, Nd);
    kcount<<<nb(nE), 256, 0, stream>>>(es, dego, nE);
    kcount<<<nb(nE), 256, 0, stream>>>(ed, degi, nE);
    krsqrt<<<nb(Ns), 256, 0, stream>>>(dego, Ns);
    krsqrt<<<nb(Nd), 256, 0, stream>>>(degi, Nd);
    kscale<<<nb((long long)Ns * HID), 256, 0, stream>>>(h[s], dego, tmp, Ns * HID);
    gemm(tmp, Wc, nullptr, xbuf, Ns, HID, HID, 0);
    kscatter<<<nb((long long)nE * 32), 256, 0, stream>>>(xbuf, es, ed, degi, hc[d], nE);
    kaddbias<<<nb((long long)Nd * HID), 256, 0, stream>>>(hc[d], bc, Nd * HID);
  };

  auto sematt = [&](float* const* zs, int K, int Nrows, const float* W1,
                    const float* b1, const float* w2, float* outp) {
    kfill<<<1, 32, 0, stream>>>(Sbuf, 0.f, 8);
    for (int k = 0; k < K; ++k) {
      gemm(zs[k], W1, b1, xbuf, Nrows, HID, HID, 1);
      ktanhdot<<<64, 256, 0, stream>>>(xbuf, w2, Sbuf, k, Nrows);
    }
    ksoftmax<<<1, 32, 0, stream>>>(Sbuf, Beta, K, 1.0f / (float)Nrows);
    kcombine<<<nb((long long)Nrows * HID), 256, 0, stream>>>(
        zs[0], (K > 1) ? zs[1] : zs[0], (K > 2) ? zs[2] : zs[0],
        (K > 3) ? zs[3] : zs[0], Beta, outp, K, Nrows * HID);
  };

  // -------- step A: input linear per ntype --------
  for (int t = 0; t < 5; ++t) {
    const float* lw = P + P_LIN + (size_t)t * 65664;
    gemm(X[t], lw, lw + 65536, h[t], NT_N[t], HID, 512, 1);
  }
  hipMemcpyAsync(drugL[0], h[1], (size_t)2000 * HID * 4, hipMemcpyDeviceToDevice, stream);
  hipMemcpyAsync(disL[0],  h[0], (size_t)3000 * HID * 4, hipMemcpyDeviceToDevice, stream);

  // -------- step B: fg1, fg2 --------
  for (int L = 0; L < 2; ++L) {
    size_t base = (L == 0) ? P_FG1 : P_FG2;
    for (int t = 0; t < 5; ++t)
      kfill<<<nb((long long)NT_N[t] * HID), 256, 0, stream>>>(hc[t], 0.f, NT_N[t] * HID);
    for (int r = 0; r < 10; ++r) {
      const float* cw = P + base + 256 + (size_t)SORTIDX[r] * 16512;
      run_rel(r, cw, cw + 16384);
    }
    for (int t = 0; t < 5; ++t)
      kbnprelu<<<nb((long long)NT_N[t] * HID), 256, 0, stream>>>(
          h[t], hc[t], P + base + 128, P + base + 0, P + base + 165376, NT_N[t] * HID);
  }
  hipMemcpyAsync(drugL[1], h[1], (size_t)2000 * HID * 4, hipMemcpyDeviceToDevice, stream);
  hipMemcpyAsync(disL[1],  h[0], (size_t)3000 * HID * 4, hipMemcpyDeviceToDevice, stream);

  // -------- step C: subnet blocks --------
  static const int BREL[5][3]  = {{0, 1, 2}, {0, 3, 4}, {4, 5, 6}, {6, 7, 8}, {8, 9, 2}};
  static const int BSLOT[5][3] = {{2, 1, 0}, {0, 1, 2}, {2, 1, 0}, {0, 1, 2}, {2, 1, 0}};
  static const int BTYPE[5][2] = {{1, 0}, {1, 4}, {4, 2}, {2, 3}, {3, 0}};
  int slot[5] = {0, 0, 0, 0, 0};
  for (int bidx = 0; bidx < 5; ++bidx) {
    size_t base = BLK_BASE[bidx];
    for (int j = 0; j < 2; ++j) {
      int t = BTYPE[bidx][j];
      kfill<<<nb((long long)NT_N[t] * HID), 256, 0, stream>>>(hc[t], 0.f, NT_N[t] * HID);
    }
    for (int j = 0; j < 3; ++j) {
      const float* cw = P + base + (size_t)BSLOT[bidx][j] * 16512;
      run_rel(BREL[bidx][j], cw, cw + 16384);
    }
    const float* pr = P + base + 49536;
    for (int j = 0; j < 2; ++j) {
      int t = BTYPE[bidx][j];
      float* dst = (slot[t] == 0) ? zA[t] : zB[t];
      kbnprelu<<<nb((long long)NT_N[t] * HID), 256, 0, stream>>>(
          dst, hc[t], nullptr, nullptr, pr, NT_N[t] * HID);
      slot[t]++;
    }
  }
  for (int t = 0; t < 5; ++t) {
    float* zs[2] = {zA[t], zB[t]};
    sematt(zs, 2, NT_N[t], P + P_SATT, P + P_SATT + 16384, P + P_SATT + 16512, h[t]);
  }
  hipMemcpyAsync(drugL[2], h[1], (size_t)2000 * HID * 4, hipMemcpyDeviceToDevice, stream);
  hipMemcpyAsync(disL[2],  h[0], (size_t)3000 * HID * 4, hipMemcpyDeviceToDevice, stream);

  // -------- step D: GAT over full graph --------
  const float* GW = P + P_GAT;
  for (int t = 0; t < 5; ++t)
    gemm(h[t], GW, nullptr, zg + (size_t)NT_OFF[t] * 256, NT_N[t], 256, HID, 1);
  keler<<<nb((long long)NTOT * 2), 256, 0, stream>>>(zg, GW + 32768, GW + 33024, el, er, NTOT);
  {
    int eoff = 0;
    for (int r = 0; r < 10; ++r) {
      kbuildedges<<<nb(RNE[r]), 256, 0, stream>>>(
          (const int*)d_in[6 + 2 * r], (const int*)d_in[7 + 2 * r], geS, geD,
          NT_OFF[RSRC[r]], NT_OFF[RDST[r]], eoff, RNE[r]);
      eoff += RNE[r];
    }
  }
  kfill<<<nb(NTOT * 2), 256, 0, stream>>>(mbuf, -1e30f, NTOT * 2);
  kfill<<<nb(NTOT * 2), 256, 0, stream>>>(den, 0.f, NTOT * 2);
  kfill<<<nb((long long)NTOT * 256), 256, 0, stream>>>(outg, 0.f, NTOT * 256);
  kedgemax<<<nb((long long)ETOT * 2), 256, 0, stream>>>(geS, geD, el, er, eattr, mbuf, ETOT);
  kedgeexp<<<nb((long long)ETOT * 2), 256, 0, stream>>>(geD, mbuf, eattr, den, ETOT);
  kedgeapply<<<nb((long long)ETOT * 64), 256, 0, stream>>>(geS, geD, eattr, den, zg, outg, ETOT);
  kgatpost<<<nb((long long)NTOT * HID), 256, 0, stream>>>(
      outg, GW + 33280, GW + 33536, GW + 33537, gout, NTOT);
  hipMemcpyAsync(disL[3],  gout,                       (size_t)3000 * HID * 4,
                 hipMemcpyDeviceToDevice, stream);
  hipMemcpyAsync(drugL[3], gout + (size_t)3000 * HID,  (size_t)2000 * HID * 4,
                 hipMemcpyDeviceToDevice, stream);

  // -------- step E: layer attention + decode --------
  sematt(drugL, 4, 2000, P + P_LATT_DRUG, P + P_LATT_DRUG + 16384,
         P + P_LATT_DRUG + 16512, Rbuf);
  sematt(disL, 4, 3000, P + P_LATT_DIS, P + P_LATT_DIS + 16384,
         P + P_LATT_DIS + 16512, Dmat);
  gemm(Dmat, P + P_DECW, nullptr, Dd, 3000, HID, HID, 1);     // D @ dec_W^T
  gemm(Rbuf, Dd, nullptr, (float*)d_out, 2000, 3000, HID, 1); // R @ (D dec_W^T)^T
}